// PhysicalModelNetwork_83958020702276
// MI455X (gfx1250) — compile-verified
//
#include <hip/hip_runtime.h>
#include <math.h>

#define WINL 512
#define STEPL 256
#define CC 257
#define NFR 128
#define NSAMP 32768
#define BROWS 128
#define OUTLEN (NFR * STEPL + WINL)   // 33280
#define CP 320                         // padded feature stride (5*64)
#define SPECP 576                      // padded spec width (9*64)
#define TWO_PI 6.28318530717958647692f

typedef __attribute__((ext_vector_type(2))) float v2f;
typedef __attribute__((ext_vector_type(8))) float v8f;

// ---------------------------------------------------------------------------
// Tiled f32 GEMM on V_WMMA_F32_16X16X4_F32.
// Block tile 64x64, 8 waves; each wave computes 16x32 (two accumulators,
// shared A fragment). K consumed in chunks of 16 staged through LDS.
// All dims pre-padded: M%64==0, N%64==0, K%16==0 -> zero divergence, EXEC
// stays all-ones around every WMMA.
// ---------------------------------------------------------------------------
__global__ __launch_bounds__(256) void gemm_wmma_tiled(
    const float* __restrict__ A, int lda,
    const float* __restrict__ B, int ldb,
    float* __restrict__ Cm, int ldc,
    int nblocks, int K) {
  int bm = blockIdx.x / nblocks;
  int bn = blockIdx.x % nblocks;
  int m0 = bm * 64, n0 = bn * 64;
  int t = threadIdx.x;
  int lane = t & 31;
  int wave = t >> 5;
  int wm = (wave & 3) * 16;      // wave row offset in block tile
  int wn = (wave >> 2) * 32;     // wave col offset in block tile

  __shared__ float Alds[64][18];   // [m][k]   row stride 18: aligned b64, no conflicts
  __shared__ float Blds[64][18];   // [n][k]   transposed B tile

  int arow = t >> 2, aseg = (t & 3) * 4;    // A tile loader: 64 rows x 16 k
  int brow = t >> 4, bseg = (t & 15) * 4;   // B tile loader: 16 k rows x 64 n

  int lm = wm + (lane & 15);
  int lncol = lane & 15;
  int kb = (lane >> 4) * 2;      // lanes 16-31 hold K+2,K+3

  v8f acc0 = {}, acc1 = {};
  for (int k0 = 0; k0 < K; k0 += 16) {
    float4 av = *(const float4*)(A + (long)(m0 + arow) * lda + k0 + aseg);
    float4 bv = *(const float4*)(B + (long)(k0 + brow) * ldb + n0 + bseg);
    __syncthreads();   // previous chunk's compute done before overwrite
    *(float4*)&Alds[arow][aseg] = av;
    Blds[bseg + 0][brow] = bv.x;
    Blds[bseg + 1][brow] = bv.y;
    Blds[bseg + 2][brow] = bv.z;
    Blds[bseg + 3][brow] = bv.w;
    __syncthreads();
#pragma unroll
    for (int kk = 0; kk < 16; kk += 4) {
      v2f a  = *(const v2f*)&Alds[lm][kk + kb];
      v2f b0 = *(const v2f*)&Blds[wn + lncol][kk + kb];
      v2f b1 = *(const v2f*)&Blds[wn + 16 + lncol][kk + kb];
      acc0 = __builtin_amdgcn_wmma_f32_16x16x4_f32(false, a, false, b0,
                                                   (short)0, acc0, false, false);
      acc1 = __builtin_amdgcn_wmma_f32_16x16x4_f32(false, a, false, b1,
                                                   (short)0, acc1, false, false);
    }
  }
  int crow = m0 + wm + ((lane >> 4) * 8);
  int ccol = n0 + wn + lncol;
#pragma unroll
  for (int i = 0; i < 8; ++i) {
    Cm[(long)(crow + i) * ldc + ccol]      = acc0[i];
    Cm[(long)(crow + i) * ldc + ccol + 16] = acc1[i];
  }
}

// Zero-pad (and copy) a weight matrix into dst (dr x dc), src is (sr x sc).
__global__ __launch_bounds__(256) void pad_weight_kernel(
    const float* __restrict__ src, float* __restrict__ dst,
    int sr, int sc, int dc, int n) {
  int i = blockIdx.x * 256 + threadIdx.x;
  if (i < n) {
    int r = i / dc, c = i % dc;
    dst[i] = (r < sr && c < sc) ? src[r * sc + c] : 0.f;
  }
}

// ---------------------------------------------------------------------------
// z = in + bias; layernorm(z)*g + be; leaky_relu(0.2). Block per row.
// Row stride CP; pad columns [257,320) are written as zero.
// ---------------------------------------------------------------------------
__global__ __launch_bounds__(256) void ln_lrelu_kernel(
    const float* __restrict__ in, const float* __restrict__ bias,
    const float* __restrict__ g, const float* __restrict__ be,
    float* __restrict__ out) {
  int r = blockIdx.x, t = threadIdx.x;
  __shared__ float red[256];
  float z0 = in[(long)r * CP + t] + bias[t];
  bool h1 = (t + 256) < CC;   // only t==0
  float z1 = h1 ? (in[(long)r * CP + t + 256] + bias[t + 256]) : 0.f;
  red[t] = z0 + z1;
  __syncthreads();
  for (int o = 128; o > 0; o >>= 1) { if (t < o) red[t] += red[t + o]; __syncthreads(); }
  float mean = red[0] * (1.f / CC);
  __syncthreads();
  float d0 = z0 - mean;
  float d1 = h1 ? (z1 - mean) : 0.f;
  red[t] = d0 * d0 + d1 * d1;
  __syncthreads();
  for (int o = 128; o > 0; o >>= 1) { if (t < o) red[t] += red[t + o]; __syncthreads(); }
  float inv = rsqrtf(red[0] * (1.f / CC) + 1e-5f);
  float y0 = d0 * inv * g[t] + be[t];
  out[(long)r * CP + t] = (y0 >= 0.f) ? y0 : 0.2f * y0;
  int c2 = t + 256;
  if (c2 < CP) {
    float y1 = 0.f;
    if (h1) {
      float yy = d1 * inv * g[c2] + be[c2];
      y1 = (yy >= 0.f) ? yy : 0.2f * yy;
    }
    out[(long)r * CP + c2] = y1;
  }
}

// t = relu(in + bf); out = t * xnorm[r] / (||t|| + 1e-8). Pads zeroed.
__global__ __launch_bounds__(256) void npnet_final_kernel(
    const float* __restrict__ in, const float* __restrict__ bf,
    const float* __restrict__ xnorm, float* __restrict__ out) {
  int r = blockIdx.x, t = threadIdx.x;
  __shared__ float red[256];
  float z0 = in[(long)r * CP + t] + bf[t];
  z0 = (z0 > 0.f) ? z0 : 0.f;
  bool h1 = (t + 256) < CC;
  float z1 = 0.f;
  if (h1) {
    z1 = in[(long)r * CP + t + 256] + bf[t + 256];
    z1 = (z1 > 0.f) ? z1 : 0.f;
  }
  red[t] = z0 * z0 + z1 * z1;
  __syncthreads();
  for (int o = 128; o > 0; o >>= 1) { if (t < o) red[t] += red[t + o]; __syncthreads(); }
  float scale = xnorm[r] / (sqrtf(red[0]) + 1e-8f);
  out[(long)r * CP + t] = z0 * scale;
  int c2 = t + 256;
  if (c2 < CP) out[(long)r * CP + c2] = h1 ? z1 * scale : 0.f;
}

// Row L2 norms over first CC cols (stride ld). One block per row.
__global__ __launch_bounds__(256) void rownorm_kernel(
    const float* __restrict__ x, float* __restrict__ nrm, int ld) {
  int r = blockIdx.x, t = threadIdx.x;
  __shared__ float red[256];
  float s = 0.f;
  for (int c = t; c < CC; c += 256) { float v = x[(long)r * ld + c]; s += v * v; }
  red[t] = s;
  __syncthreads();
  for (int o = 128; o > 0; o >>= 1) { if (t < o) red[t] += red[t + o]; __syncthreads(); }
  if (t == 0) nrm[r] = sqrtf(red[0]);
}

// ---------------------------------------------------------------------------
// Window + 512-pt real DFT magnitude. One block per (row, frame).
// Exact twiddles via 512-entry LDS cosine table; sin(j)=cos((j+384)&511).
// Output row stride CP, pad cols zeroed.
// ---------------------------------------------------------------------------
__global__ __launch_bounds__(256) void window_dft_kernel(
    const float* __restrict__ x, float* __restrict__ mag) {
  int bid = blockIdx.x;             // r*NFR + f
  int r = bid >> 7, f = bid & 127;
  int t = threadIdx.x;
  __shared__ float win[WINL];
  __shared__ float ct[WINL];
  for (int w = t; w < WINL; w += 256) {
    float c = cosf((float)w * (TWO_PI / (float)WINL));
    ct[w] = c;
    int gi = f * STEPL + w;
    float v = (gi < NSAMP) ? x[(long)r * NSAMP + gi] : 0.f;  // zero pad tail
    win[w] = v * (0.5f - 0.5f * c);                          // Hann
  }
  __syncthreads();
  for (int k = t; k < CP; k += 256) {
    float o = 0.f;
    if (k < CC) {
      float re = 0.f, im = 0.f;
      for (int n = 0; n < WINL; ++n) {
        int idx = (k * n) & 511;
        float w = win[n];
        re += w * ct[idx];
        im -= w * ct[(idx + 384) & 511];
      }
      o = sqrtf(re * re + im * im);
    }
    mag[(long)bid * CP + k] = o;
  }
}

// Per scan step: phase update, irFFT(512), Hann, overlap-add. Block per row.
__global__ __launch_bounds__(256) void synth_kernel(
    const float* __restrict__ spec, float* __restrict__ phase,
    float* __restrict__ outac, int f) {
  int r = blockIdx.x, t = threadIdx.x;
  __shared__ float R[CC];
  __shared__ float I[CC];
  __shared__ float ct[WINL];
  for (int w = t; w < WINL; w += 256)
    ct[w] = cosf((float)w * (TWO_PI / (float)WINL));
  for (int k = t; k < CC; k += 256) {
    float re = spec[(long)r * SPECP + k];
    float im = spec[(long)r * SPECP + CC + k];
    float gd = 3.14159265358979323846f * (float)k / 256.f;   // linspace(0,pi,257)
    float ph = phase[(long)r * CC + k] + gd * (tanhf(im) * 0.1f);
    phase[(long)r * CC + k] = ph;
    R[k] = re * cosf(ph);
    I[k] = re * sinf(ph);
  }
  __syncthreads();
  for (int n = t; n < WINL; n += 256) {
    float acc = R[0] + ((n & 1) ? -R[256] : R[256]);
    for (int k = 1; k < 256; ++k) {
      int idx = (k * n) & 511;
      acc += 2.f * (R[k] * ct[idx] - I[k] * ct[(idx + 384) & 511]);
    }
    float smp = acc * (1.f / (float)WINL);
    float hann = 0.5f - 0.5f * ct[n];
    outac[(long)r * OUTLEN + f * STEPL + n] += smp * hann;
  }
}

// ---------------------------------------------------------------------------
// Small elementwise kernels (row stride CP unless noted)
// ---------------------------------------------------------------------------
__global__ __launch_bounds__(256) void zero_kernel(float* p, int n) {
  int i = blockIdx.x * 256 + threadIdx.x;
  if (i < n) p[i] = 0.f;
}

__global__ __launch_bounds__(256) void add_bias_kernel(float* x, const float* __restrict__ b, int n) {
  int i = blockIdx.x * 256 + threadIdx.x;
  if (i < n) {
    int c = i % CP;
    if (c < CC) x[i] += b[c];
  }
}

__global__ __launch_bounds__(256) void add2_kernel(const float* __restrict__ a,
                                                   const float* __restrict__ b,
                                                   float* __restrict__ o, int n) {
  int i = blockIdx.x * 256 + threadIdx.x;
  if (i < n) o[i] = a[i] + b[i];
}

// h0[r,c] = hidden[r,c] + control[(r*NFR + f), c]  (both stride CP)
__global__ __launch_bounds__(256) void add_ctrl_kernel(const float* __restrict__ hidden,
                                                       const float* __restrict__ control,
                                                       float* __restrict__ h0, int f) {
  int i = blockIdx.x * 256 + threadIdx.x;
  if (i < BROWS * CP) {
    int r = i / CP, c = i % CP;
    h0[i] = (c < CC) ? hidden[i] + control[((long)r * NFR + f) * CP + c] : 0.f;
  }
}

// lr = 0.2*sigmoid(lkraw + bf); leaked = hn*lr; hidden(carry) = hn - leaked
__global__ __launch_bounds__(256) void leak_update_kernel(
    const float* __restrict__ hn, const float* __restrict__ lkraw,
    const float* __restrict__ bf, float* __restrict__ leaked,
    float* __restrict__ hidden) {
  int i = blockIdx.x * 256 + threadIdx.x;
  if (i < BROWS * CP) {
    int c = i % CP;
    if (c < CC) {
      float v = lkraw[i] + bf[c];
      float lr = 0.2f / (1.f + expf(-v));
      float l = hn[i] * lr;
      leaked[i] = l;
      hidden[i] = hn[i] - l;
    } else {
      leaked[i] = 0.f;
      hidden[i] = 0.f;
    }
  }
}

__global__ __launch_bounds__(256) void copy_out_kernel(const float* __restrict__ outac,
                                                       float* __restrict__ out) {
  int i = blockIdx.x * 256 + threadIdx.x;
  if (i < BROWS * NSAMP) {
    int r = i / NSAMP, c = i % NSAMP;
    out[i] = outac[(long)r * OUTLEN + c];
  }
}

// ---------------------------------------------------------------------------
// Host orchestration
// ---------------------------------------------------------------------------
extern "C" void kernel_launch(void* const* d_in, const int* in_sizes, int n_in,
                              void* d_out, int out_size, void* d_ws, size_t ws_size,
                              hipStream_t stream) {
  (void)in_sizes; (void)n_in; (void)out_size; (void)ws_size;
  const float* imp        = (const float*)d_in[0];
  const float* embeddings = (const float*)d_in[1];
  const float* proj_W     = (const float*)d_in[2];
  const float* proj_b     = (const float*)d_in[3];
  const float* spec_W     = (const float*)d_in[4];
  const float* W_[4]  = {(const float*)d_in[5],  (const float*)d_in[11],
                         (const float*)d_in[17], (const float*)d_in[23]};
  const float* b_[4]  = {(const float*)d_in[6],  (const float*)d_in[12],
                         (const float*)d_in[18], (const float*)d_in[24]};
  const float* g_[4]  = {(const float*)d_in[7],  (const float*)d_in[13],
                         (const float*)d_in[19], (const float*)d_in[25]};
  const float* be_[4] = {(const float*)d_in[8],  (const float*)d_in[14],
                         (const float*)d_in[20], (const float*)d_in[26]};
  const float* Wf_[4] = {(const float*)d_in[9],  (const float*)d_in[15],
                         (const float*)d_in[21], (const float*)d_in[27]};
  const float* bf_[4] = {(const float*)d_in[10], (const float*)d_in[16],
                         (const float*)d_in[22], (const float*)d_in[28]};
  // prefix order: 0=embed, 1=transform, 2=eshape, 3=leak

  float* ws = (float*)d_ws;
  const long BIG = (long)16384 * CP;           // 5,242,880
  float* magbuf    = ws;                       // mag, later reused as control
  float* T0        = magbuf + BIG;
  float* T1        = T0 + BIG;
  float* outac     = T1 + BIG;                 // 128*33280
  float* embp      = outac + (long)BROWS * OUTLEN;
  float* shape_lat = embp + BROWS * CP;
  float* hidden    = shape_lat + BROWS * CP;
  float* h0        = hidden + BROWS * CP;
  float* hn        = h0 + BROWS * CP;
  float* s0        = hn + BROWS * CP;
  float* s1        = s0 + BROWS * CP;
  float* leaked    = s1 + BROWS * CP;
  float* phase     = leaked + BROWS * CP;      // 128*257 (dense)
  float* spec      = phase + BROWS * CC + 48;  // keep 16-float alignment
  float* nrmB      = spec + BROWS * SPECP;     // 16384
  float* nrmS      = nrmB + 16384;             // 128
  float* wpad      = nrmS + 128;               // padded weights region
  // padded weights: per prefix: 3 layer W (320x320) + Wf (320x320)
  const int SQ = CP * CP;                      // 102400
  float* WP[4], *WfP[4];
  float* wp = wpad;
  for (int p = 0; p < 4; ++p) { WP[p] = wp; wp += 3 * SQ; WfP[p] = wp; wp += SQ; }
  float* projP = wp; wp += 256 * CP;           // 256x320
  float* specP = wp; wp += CP * SPECP;         // 320x576

  auto gemm = [&](const float* A, int lda, const float* B, int ldb,
                  float* Cp, int ldc, int M, int Npad, int K) {
    int grid = (M / 64) * (Npad / 64);
    gemm_wmma_tiled<<<grid, 256, 0, stream>>>(A, lda, B, ldb, Cp, ldc, Npad / 64, K);
  };
  auto pad = [&](const float* src, float* dst, int sr, int sc, int dr, int dc) {
    int n = dr * dc;
    pad_weight_kernel<<<(n + 255) / 256, 256, 0, stream>>>(src, dst, sr, sc, dc, n);
  };
  auto npnet = [&](const float* X, int p, const float* norms,
                   float* t0, float* t1, float* out, int M) {
    gemm(X, CP, WP[p], CP, t0, CP, M, CP, CP);
    ln_lrelu_kernel<<<M, 256, 0, stream>>>(t0, b_[p], g_[p], be_[p], t1);
    gemm(t1, CP, WP[p] + SQ, CP, t0, CP, M, CP, CP);
    ln_lrelu_kernel<<<M, 256, 0, stream>>>(t0, b_[p] + CC, g_[p] + CC, be_[p] + CC, t1);
    gemm(t1, CP, WP[p] + 2 * SQ, CP, t0, CP, M, CP, CP);
    ln_lrelu_kernel<<<M, 256, 0, stream>>>(t0, b_[p] + 2 * CC, g_[p] + 2 * CC, be_[p] + 2 * CC, t1);
    gemm(t1, CP, WfP[p], CP, t0, CP, M, CP, CP);
    npnet_final_kernel<<<M, 256, 0, stream>>>(t0, bf_[p], norms, out);
  };

  // ---- pad all weights into workspace (once per call; deterministic) ----
  for (int p = 0; p < 4; ++p) {
    for (int l = 0; l < 3; ++l) pad(W_[p] + l * CC * CC, WP[p] + l * SQ, CC, CC, CP, CP);
    pad(Wf_[p], WfP[p], CC, CC, CP, CP);
  }
  pad(proj_W, projP, 256, CC, 256, CP);
  pad(spec_W, specP, CC, 2 * CC, CP, SPECP);

  // ---- init state ----
  zero_kernel<<<(BROWS * OUTLEN + 255) / 256, 256, 0, stream>>>(outac, BROWS * OUTLEN);
  zero_kernel<<<(BROWS * CP + 255) / 256, 256, 0, stream>>>(hidden, BROWS * CP);
  zero_kernel<<<(BROWS * CC + 255) / 256, 256, 0, stream>>>(phase, BROWS * CC);

  // ---- STFT magnitudes ----
  window_dft_kernel<<<16384, 256, 0, stream>>>(imp, magbuf);
  rownorm_kernel<<<16384, 256, 0, stream>>>(magbuf, nrmB, CP);

  // ---- control = npnet(mag) over all 16384 frame-vectors ----
  npnet(magbuf, 0, nrmB, T0, T1, magbuf, 16384);

  // ---- embedding projection + shape latent ----
  gemm(embeddings, 256, projP, CP, embp, CP, BROWS, CP, 256);
  add_bias_kernel<<<(BROWS * CP + 255) / 256, 256, 0, stream>>>(embp, proj_b, BROWS * CP);
  rownorm_kernel<<<BROWS, 256, 0, stream>>>(embp, nrmS, CP);
  npnet(embp, 2, nrmS, s0, s1, shape_lat, BROWS);

  // ---- sequential scan over 128 frames ----
  const int EW = (BROWS * CP + 255) / 256;
  for (int f = 0; f < NFR; ++f) {
    add_ctrl_kernel<<<EW, 256, 0, stream>>>(hidden, magbuf, h0, f);
    rownorm_kernel<<<BROWS, 256, 0, stream>>>(h0, nrmS, CP);
    npnet(h0, 1, nrmS, s0, s1, hn, BROWS);

    add2_kernel<<<EW, 256, 0, stream>>>(hn, shape_lat, s0, BROWS * CP);
    gemm(s0, CP, WP[3], CP, s1, CP, BROWS, CP, CP);
    ln_lrelu_kernel<<<BROWS, 256, 0, stream>>>(s1, b_[3], g_[3], be_[3], s0);
    gemm(s0, CP, WP[3] + SQ, CP, s1, CP, BROWS, CP, CP);
    ln_lrelu_kernel<<<BROWS, 256, 0, stream>>>(s1, b_[3] + CC, g_[3] + CC, be_[3] + CC, s0);
    gemm(s0, CP, WP[3] + 2 * SQ, CP, s1, CP, BROWS, CP, CP);
    ln_lrelu_kernel<<<BROWS, 256, 0, stream>>>(s1, b_[3] + 2 * CC, g_[3] + 2 * CC, be_[3] + 2 * CC, s0);
    gemm(s0, CP, WfP[3], CP, s1, CP, BROWS, CP, CP);   // pre-bias leak logits

    leak_update_kernel<<<EW, 256, 0, stream>>>(hn, s1, bf_[3], leaked, hidden);
    gemm(leaked, CP, specP, SPECP, spec, SPECP, BROWS, SPECP, CP);
    synth_kernel<<<BROWS, 256, 0, stream>>>(spec, phase, outac, f);
  }

  copy_out_kernel<<<(BROWS * NSAMP + 255) / 256, 256, 0, stream>>>(outac, (float*)d_out);
}